// myModel_82738249990227
// MI455X (gfx1250) — compile-verified
//
#include <hip/hip_runtime.h>
#include <hip/hip_bf16.h>

typedef __attribute__((ext_vector_type(2))) float v2f;
typedef __attribute__((ext_vector_type(8))) float v8f;

#define NBATCH 16
#define CIN    10
#define HH     512
#define WW     512
#define HWSZ   (HH*WW)          // 262144 = 2^18
#define NPIX   (NBATCH*HWSZ)    // 4194304
#define SLICES 16
#define EPSBN  1e-5f

__device__ __forceinline__ float sigmoidf(float v){ return 1.0f/(1.0f + __expf(-v)); }

#if __has_builtin(__builtin_amdgcn_wmma_f32_16x16x4_f32)
#define HAVE_WMMA_F32X4 1
#endif

// One K=4 step of a 16x16 f32 GEMM striped across the wave.
// A(16x4): row = lane&15; VGPR0/1 hold K = k0+2*(lane>>4) .. +1
// B(4x16): col = lane&15; VGPR0/1 hold K = k0+2*(lane>>4) .. +1
// C/D(16x16): VGPR j -> row j+8*(lane>>4), col = lane&15
__device__ __forceinline__ v8f wmma_k4(float a0, float a1, float b0, float b1, v8f acc){
#ifdef HAVE_WMMA_F32X4
  v2f av; av.x = a0; av.y = a1;
  v2f bv; bv.x = b0; bv.y = b1;
  return __builtin_amdgcn_wmma_f32_16x16x4_f32(false, av, false, bv, (short)0, acc, false, false);
#else
  acc[0] = fmaf(a0, b0, fmaf(a1, b1, acc[0]));  // compile-keepalive fallback
  return acc;
#endif
}

// ---- K1: 1x1 conv 10->16 (WMMA) + BN1 channel sums -------------------------
__global__ __launch_bounds__(256) void k_bandconv1(
    const float* __restrict__ x, const float* __restrict__ w, const float* __restrict__ b,
    float* __restrict__ y, float* __restrict__ gsum, float* __restrict__ gsq)
{
  __shared__ float ssum[16], ssq[16];
  int lt = threadIdx.x;
  if (lt < 16){ ssum[lt] = 0.f; ssq[lt] = 0.f; }
  __syncthreads();

  int lane = lt & 31, wv = lt >> 5;
  int m = lane & 15, half = lane >> 4;
  long q   = ((long)blockIdx.x * 8 + wv) * 16 + m;   // pixel id for this lane's column
  int  n   = (int)(q >> 18);
  int  off = (int)(q & (HWSZ - 1));
  const float* xs = x + ((long)n * CIN) * HWSZ + off;

  v8f acc = {};
#pragma unroll
  for (int k0 = 0; k0 < 12; k0 += 4){               // K=10 padded to 12
    int ka = k0 + half * 2, kb = ka + 1;
    float a0 = (ka < CIN) ? w[m * CIN + ka] : 0.f;
    float a1 = (kb < CIN) ? w[m * CIN + kb] : 0.f;
    float b0 = (ka < CIN) ? xs[(long)ka * HWSZ] : 0.f;
    float b1 = (kb < CIN) ? xs[(long)kb * HWSZ] : 0.f;
    acc = wmma_k4(a0, a1, b0, b1, acc);
  }

  float* ys = y + ((long)n * 16) * HWSZ + off;
#pragma unroll
  for (int j = 0; j < 8; ++j){
    int oc = j + half * 8;
    float v = acc[j] + b[oc];
    ys[(long)oc * HWSZ] = v;
    atomicAdd(&ssum[oc], v);
    atomicAdd(&ssq[oc], v * v);
  }
  __syncthreads();
  if (lt < 16){ atomicAdd(&gsum[lt], ssum[lt]); atomicAdd(&gsq[lt], ssq[lt]); }
}

// ---- BN finalize: scale/shift from sums ------------------------------------
__global__ void k_bn_finalize(const float* sum, const float* sq,
                              const float* g, const float* be,
                              float cnt, int C, float* scale, float* shift)
{
  int c = threadIdx.x;
  if (c >= C) return;
  float mean = sum[c] / cnt;
  float var  = sq[c] / cnt - mean * mean;
  float r = rsqrtf(var + EPSBN);
  float s = g[c] * r;
  scale[c] = s;
  shift[c] = be[c] - mean * s;
}

// ---- K3: sigmoid(BN1) on load, 1x1 conv 16->16 (WMMA), in place, BN2 sums --
__global__ __launch_bounds__(256) void k_bandconv2(
    float* __restrict__ y, const float* __restrict__ w, const float* __restrict__ b,
    const float* __restrict__ sc1, const float* __restrict__ sh1,
    float* __restrict__ gsum, float* __restrict__ gsq)
{
  __shared__ float ssum[16], ssq[16];
  int lt = threadIdx.x;
  if (lt < 16){ ssum[lt] = 0.f; ssq[lt] = 0.f; }
  __syncthreads();

  int lane = lt & 31, wv = lt >> 5;
  int m = lane & 15, half = lane >> 4;
  long q   = ((long)blockIdx.x * 8 + wv) * 16 + m;
  int  n   = (int)(q >> 18);
  int  off = (int)(q & (HWSZ - 1));
  float* ys = y + ((long)n * 16) * HWSZ + off;

  v8f acc = {};
#pragma unroll
  for (int k0 = 0; k0 < 16; k0 += 4){
    int ka = k0 + half * 2, kb = ka + 1;
    float a0 = w[m * 16 + ka];
    float a1 = w[m * 16 + kb];
    float b0 = sigmoidf(sc1[ka] * ys[(long)ka * HWSZ] + sh1[ka]);
    float b1 = sigmoidf(sc1[kb] * ys[(long)kb * HWSZ] + sh1[kb]);
    acc = wmma_k4(a0, a1, b0, b1, acc);
  }
  // In-place store is safe: the wave owns this full 16ch x 16px tile, and the
  // stores depend (through the WMMA chain) on every load above.
#pragma unroll
  for (int j = 0; j < 8; ++j){
    int oc = j + half * 8;
    float v = acc[j] + b[oc];
    ys[(long)oc * HWSZ] = v;
    atomicAdd(&ssum[oc], v);
    atomicAdd(&ssq[oc], v * v);
  }
  __syncthreads();
  if (lt < 16){ atomicAdd(&gsum[lt], ssum[lt]); atomicAdd(&gsq[lt], ssq[lt]); }
}

// ---- K5: h = sigmoid(BN2) in place + per-(n,c) sum/sumsq -------------------
__global__ __launch_bounds__(256) void k_hfinal_stats(
    float* __restrict__ y, const float* sc2, const float* sh2,
    float* __restrict__ psum, float* __restrict__ psq)
{
  __shared__ float ls, lq;
  if (threadIdx.x == 0){ ls = 0.f; lq = 0.f; }
  __syncthreads();
  int blk = blockIdx.x;                 // NBATCH*16*SLICES blocks
  int sl  = blk & (SLICES - 1);
  int c   = (blk >> 4) & 15;
  int n   = blk >> 8;
  float scl = sc2[c], sft = sh2[c];
  float* p = y + ((long)(n * 16 + c)) * HWSZ + sl * (HWSZ / SLICES);
  float s = 0.f, qq = 0.f;
  for (int i = threadIdx.x; i < HWSZ / SLICES; i += 256){
    float h = sigmoidf(scl * p[i] + sft);
    p[i] = h;
    s += h; qq += h * h;
  }
  atomicAdd(&ls, s); atomicAdd(&lq, qq);
  __syncthreads();
  if (threadIdx.x == 0){ atomicAdd(&psum[n * 16 + c], ls); atomicAdd(&psq[n * 16 + c], lq); }
}

// ---- K6: per-sample top-3 channels by unbiased variance (stable ties) ------
__global__ void k_topk(const float* psum, const float* psq, int* idx)
{
  int n = threadIdx.x;
  if (n >= NBATCH) return;
  float v[16]; bool used[16];
  for (int c = 0; c < 16; ++c){
    float s = psum[n * 16 + c], q = psq[n * 16 + c];
    v[c] = (q - s * s / (float)HWSZ) / (float)(HWSZ - 1);
    used[c] = false;
  }
  for (int j = 0; j < 3; ++j){
    int best = 0; float bv = -3.4e38f;
    for (int c = 0; c < 16; ++c)
      if (!used[c] && v[c] > bv){ bv = v[c]; best = c; }
    used[best] = true;
    idx[n * 3 + j] = best;
  }
}

// ---- K7/K8: 3x3 conv (pad 1) + sigmoid + 2x2 maxpool -----------------------
__global__ __launch_bounds__(256) void k_conv3x3_pool(
    const float* __restrict__ in, const float* __restrict__ wgt, const float* __restrict__ bias,
    const int* __restrict__ sel, float* __restrict__ outp,
    int ICH, int CI, int CO, int IH, int IW)
{
  int OH = IH >> 1, OW = IW >> 1;
  long t = (long)blockIdx.x * 256 + threadIdx.x;
  int ox = (int)(t % OW); long r = t / OW;
  int oy = (int)(r % OH); r /= OH;
  int co = (int)(r % CO);
  int n  = (int)(r / CO);

  float a00, a01, a10, a11;
  a00 = a01 = a10 = a11 = bias[co];
  int iy0 = 2 * oy - 1, ix0 = 2 * ox - 1;
  for (int ci = 0; ci < CI; ++ci){
    int ch = sel ? sel[n * 3 + ci] : ci;
    const float* ip = in + ((long)(n * ICH + ch)) * IH * IW;
    float p[4][4];
    for (int rr = 0; rr < 4; ++rr){
      int iy = iy0 + rr;
      for (int cc = 0; cc < 4; ++cc){
        int ix = ix0 + cc;
        p[rr][cc] = (iy >= 0 && iy < IH && ix >= 0 && ix < IW) ? ip[(long)iy * IW + ix] : 0.f;
      }
    }
    const float* wp = wgt + ((long)(co * CI + ci)) * 9;
    for (int ky = 0; ky < 3; ++ky)
      for (int kx = 0; kx < 3; ++kx){
        float wvv = wp[ky * 3 + kx];
        a00 = fmaf(wvv, p[ky][kx],       a00);
        a01 = fmaf(wvv, p[ky][kx + 1],   a01);
        a10 = fmaf(wvv, p[ky + 1][kx],   a10);
        a11 = fmaf(wvv, p[ky + 1][kx + 1], a11);
      }
  }
  float mx = fmaxf(fmaxf(sigmoidf(a00), sigmoidf(a01)),
                   fmaxf(sigmoidf(a10), sigmoidf(a11)));
  outp[((long)(n * CO + co) * OH + oy) * OW + ox] = mx;
}

// ---- bilinear helpers (half-pixel centers, edge clamp) ---------------------
struct BiC { int a, b; float t; };
__device__ __forceinline__ BiC bicoord(int o, int s, int nsz){
  float f  = ((float)o + 0.5f) / (float)s - 0.5f;
  float fl = floorf(f);
  int i0 = (int)fl;
  BiC r;
  r.t = f - fl;
  r.a = min(max(i0, 0), nsz - 1);
  r.b = min(max(i0 + 1, 0), nsz - 1);
  return r;
}
__device__ __forceinline__ float bilerp(const float* p, int Wd, BiC y, BiC x){
  float v0 = p[(long)y.a * Wd + x.a] * (1.f - x.t) + p[(long)y.a * Wd + x.b] * x.t;
  float v1 = p[(long)y.b * Wd + x.a] * (1.f - x.t) + p[(long)y.b * Wd + x.b] * x.t;
  return v0 * (1.f - y.t) + v1 * y.t;
}

// ---- K9: channel stats of relu(upsample(x1,2)) and relu(upsample(x2,4)) ----
__global__ __launch_bounds__(256) void k_upstats(
    const float* __restrict__ x1, const float* __restrict__ x2,
    float* u1s, float* u1q, float* u2s, float* u2q)
{
  __shared__ float s1s[8], s1qv[8], s2s[16], s2qv[16];
  int lt = threadIdx.x;
  if (lt < 8){ s1s[lt] = 0.f; s1qv[lt] = 0.f; }
  if (lt < 16){ s2s[lt] = 0.f; s2qv[lt] = 0.f; }
  __syncthreads();

  long t = (long)blockIdx.x * 256 + lt;
  int xo = (int)(t & 511), yo = (int)((t >> 9) & 511), n = (int)(t >> 18);

  BiC y2 = bicoord(yo, 4, 128), x2c = bicoord(xo, 4, 128);
  const float* p2 = x2 + (long)n * 16 * 16384;
  for (int c = 0; c < 16; ++c){
    float rr = fmaxf(bilerp(p2 + (long)c * 16384, 128, y2, x2c), 0.f);
    atomicAdd(&s2s[c], rr); atomicAdd(&s2qv[c], rr * rr);
  }
  BiC y1 = bicoord(yo, 2, 256), x1c = bicoord(xo, 2, 256);
  const float* p1 = x1 + (long)n * 8 * 65536;
  for (int c = 0; c < 8; ++c){
    float rr = fmaxf(bilerp(p1 + (long)c * 65536, 256, y1, x1c), 0.f);
    atomicAdd(&s1s[c], rr); atomicAdd(&s1qv[c], rr * rr);
  }
  __syncthreads();
  if (lt < 16){ atomicAdd(&u2s[lt], s2s[lt]); atomicAdd(&u2q[lt], s2qv[lt]); }
  if (lt < 8){ atomicAdd(&u1s[lt], s1s[lt]); atomicAdd(&u1q[lt], s1qv[lt]); }
}

// ---- K11: fused upsample+relu+BN+concat + 1x1 conv 24->5 -------------------
__global__ __launch_bounds__(256) void k_final(
    const float* __restrict__ x1, const float* __restrict__ x2,
    const float* sc3, const float* sh3, const float* sc4, const float* sh4,
    const float* __restrict__ wk, const float* __restrict__ bk,
    float* __restrict__ outp)
{
  long t = (long)blockIdx.x * 256 + threadIdx.x;
  int xo = (int)(t & 511), yo = (int)((t >> 9) & 511), n = (int)(t >> 18);

  float f1[8], f2[16];
  BiC y1 = bicoord(yo, 2, 256), x1c = bicoord(xo, 2, 256);
  const float* p1 = x1 + (long)n * 8 * 65536;
  for (int c = 0; c < 8; ++c){
    float v = bilerp(p1 + (long)c * 65536, 256, y1, x1c);
    f1[c] = sc4[c] * fmaxf(v, 0.f) + sh4[c];
  }
  BiC y2 = bicoord(yo, 4, 128), x2c = bicoord(xo, 4, 128);
  const float* p2 = x2 + (long)n * 16 * 16384;
  for (int c = 0; c < 16; ++c){
    float v = bilerp(p2 + (long)c * 16384, 128, y2, x2c);
    f2[c] = sc3[c] * fmaxf(v, 0.f) + sh3[c];
  }
  long pix = ((long)n * 5) * HWSZ + (long)yo * WW + xo;
  for (int c = 0; c < 5; ++c){
    float o = bk[c];
    for (int j = 0; j < 8;  ++j) o = fmaf(wk[c * 24 + j],     f1[j], o);
    for (int j = 0; j < 16; ++j) o = fmaf(wk[c * 24 + 8 + j], f2[j], o);
    outp[pix + (long)c * HWSZ] = o;
  }
}

// ---------------------------------------------------------------------------
extern "C" void kernel_launch(void* const* d_in, const int* in_sizes, int n_in,
                              void* d_out, int out_size, void* d_ws, size_t ws_size,
                              hipStream_t stream)
{
  (void)in_sizes; (void)n_in; (void)out_size; (void)ws_size;
  const float* x       = (const float*)d_in[0];
  const float* w_ext1  = (const float*)d_in[1];
  const float* b_ext1  = (const float*)d_in[2];
  const float* g_ebn1  = (const float*)d_in[3];
  const float* be_ebn1 = (const float*)d_in[4];
  const float* w_ext2  = (const float*)d_in[5];
  const float* b_ext2  = (const float*)d_in[6];
  const float* g_ebn2  = (const float*)d_in[7];
  const float* be_ebn2 = (const float*)d_in[8];
  const float* w_c1    = (const float*)d_in[9];
  const float* b_c1    = (const float*)d_in[10];
  const float* w_c2    = (const float*)d_in[11];
  const float* b_c2    = (const float*)d_in[12];
  const float* g_bn1   = (const float*)d_in[13];
  const float* be_bn1  = (const float*)d_in[14];
  const float* g_bn2   = (const float*)d_in[15];
  const float* be_bn2  = (const float*)d_in[16];
  const float* w_1k    = (const float*)d_in[17];
  const float* b_1k    = (const float*)d_in[18];
  float* outp = (float*)d_out;

  char* ws = (char*)d_ws;
  size_t offH  = 0;
  size_t offX1 = offH  + (size_t)NBATCH * 16 * HWSZ * 4;       // 268 MB h/y buffer
  size_t offX2 = offX1 + (size_t)NBATCH * 8 * 256 * 256 * 4;   // 33.5 MB x1
  size_t offSt = offX2 + (size_t)NBATCH * 16 * 128 * 128 * 4;  // 16.8 MB x2
  float* bufH = (float*)(ws + offH);
  float* x1b  = (float*)(ws + offX1);
  float* x2b  = (float*)(ws + offX2);
  float* st   = (float*)(ws + offSt);

  float* s1sum = st;        float* s1sq = st + 16;
  float* sc1   = st + 32;   float* sh1  = st + 48;
  float* s2sum = st + 64;   float* s2sq = st + 80;
  float* sc2   = st + 96;   float* sh2  = st + 112;
  float* psum  = st + 128;  float* psq  = st + 384;   // 16x16 each
  int*   topidx = (int*)(st + 640);                   // 16x3
  float* u2s = st + 688; float* u2q = st + 704;
  float* u1s = st + 720; float* u1q = st + 728;
  float* sc3 = st + 736; float* sh3 = st + 752;
  float* sc4 = st + 768; float* sh4 = st + 776;       // ends at st+784

  hipMemsetAsync(st, 0, 784 * sizeof(float), stream); // zero accumulators each call

  const float CNT = (float)((long)NBATCH * HWSZ);     // all BNs reduce over N*H*W

  k_bandconv1<<<NPIX / (16 * 8), 256, 0, stream>>>(x, w_ext1, b_ext1, bufH, s1sum, s1sq);
  k_bn_finalize<<<1, 16, 0, stream>>>(s1sum, s1sq, g_ebn1, be_ebn1, CNT, 16, sc1, sh1);
  k_bandconv2<<<NPIX / (16 * 8), 256, 0, stream>>>(bufH, w_ext2, b_ext2, sc1, sh1, s2sum, s2sq);
  k_bn_finalize<<<1, 16, 0, stream>>>(s2sum, s2sq, g_ebn2, be_ebn2, CNT, 16, sc2, sh2);
  k_hfinal_stats<<<NBATCH * 16 * SLICES, 256, 0, stream>>>(bufH, sc2, sh2, psum, psq);
  k_topk<<<1, 16, 0, stream>>>(psum, psq, topidx);
  k_conv3x3_pool<<<(NBATCH * 8 * 256 * 256) / 256, 256, 0, stream>>>(
      bufH, w_c1, b_c1, topidx, x1b, 16, 3, 8, 512, 512);
  k_conv3x3_pool<<<(NBATCH * 16 * 128 * 128) / 256, 256, 0, stream>>>(
      x1b, w_c2, b_c2, nullptr, x2b, 8, 8, 16, 256, 256);
  k_upstats<<<(NBATCH * HWSZ) / 256, 256, 0, stream>>>(x1b, x2b, u1s, u1q, u2s, u2q);
  k_bn_finalize<<<1, 16, 0, stream>>>(u2s, u2q, g_bn1, be_bn1, CNT, 16, sc3, sh3);
  k_bn_finalize<<<1, 8, 0, stream>>>(u1s, u1q, g_bn2, be_bn2, CNT, 8, sc4, sh4);
  k_final<<<(NBATCH * HWSZ) / 256, 256, 0, stream>>>(
      x1b, x2b, sc3, sh3, sc4, sh4, w_1k, b_1k, outp);
}